// MyGCN_88149908783507
// MI455X (gfx1250) — compile-verified
//
#include <hip/hip_runtime.h>
#include <hip/hip_bf16.h>

// ---------------------------------------------------------------------------
// GCN (3x GCNConv + mean-pool + linear) for MI455X (gfx1250, wave32, WMMA).
// Dense parts use V_WMMA_F32_16X16X4_F32 (f32 in/out, no precision loss).
// Sparse scatter is L2-resident gather + global f32 atomics.
// ---------------------------------------------------------------------------

typedef __attribute__((ext_vector_type(2))) float v2f;
typedef __attribute__((ext_vector_type(8))) float v8f;

#define N_NODES  100000
#define N_EDGES  1600000
#define N_GRAPHS 256
#define F_IN     128
#define HID      64

// -------------------------------------------------------------------
// Degree / normalization:  deg[i] = 1 (self loop) + #edges with dst==i
// dinv[i] = rsqrt(deg[i])
// -------------------------------------------------------------------
__global__ void init_deg_kernel(float* __restrict__ deg, int n) {
    int i = blockIdx.x * blockDim.x + threadIdx.x;
    if (i < n) deg[i] = 1.0f;   // self loop
}

__global__ void deg_count_kernel(const int* __restrict__ dst, float* __restrict__ deg, int e) {
    int i = blockIdx.x * blockDim.x + threadIdx.x;
    if (i < e) atomicAdd(&deg[dst[i]], 1.0f);
}

__global__ void dinv_kernel(const float* __restrict__ deg, float* __restrict__ dinv, int n) {
    int i = blockIdx.x * blockDim.x + threadIdx.x;
    if (i < n) dinv[i] = __frsqrt_rn(deg[i]);   // deg >= 1 always
}

// -------------------------------------------------------------------
// GEMM: xw = A[M,K] @ W[K,64] via V_WMMA_F32_16X16X4_F32.
// One wave -> one 16x16 tile; 4 waves/block cover the 64-wide output.
// Fused epilogue: out[i][:] = dinv[i]^2 * xw[i][:]  (self-loop term),
// so the edge-scatter kernel only accumulates real edges on top.
// -------------------------------------------------------------------
__global__ void gemm_selfloop_kernel(const float* __restrict__ A,
                                     const float* __restrict__ W,
                                     const float* __restrict__ dinv,
                                     float* __restrict__ xw,
                                     float* __restrict__ out,
                                     int M, int K) {
    const int lane = threadIdx.x & 31;
    const int wave = threadIdx.x >> 5;       // 0..3 -> 16-col tile
    const int half = lane >> 4;              // 0 | 1
    const int m    = lane & 15;
    const int row0 = blockIdx.x * 16;
    const int row  = row0 + m;
    const int rowc = (row < M) ? row : (M - 1);   // clamp loads; EXEC stays full for WMMA
    const int col  = wave * 16 + m;

    v8f acc = {};
    const float* arow = A + (long long)rowc * K;
    for (int k = 0; k < K; k += 4) {
        const int ka = k + half * 2;
        v2f a, b;
        a.x = arow[ka];
        a.y = arow[ka + 1];
        b.x = W[(long long)ka * HID + col];
        b.y = W[(long long)(ka + 1) * HID + col];
        acc = __builtin_amdgcn_wmma_f32_16x16x4_f32(
                  false, a, false, b, (short)0, acc, false, false);
    }

    #pragma unroll
    for (int r = 0; r < 8; ++r) {
        const int rr = row0 + r + 8 * half;
        if (rr < M) {
            const float v  = acc[r];
            const float di = dinv[rr];
            xw[(long long)rr * HID + col]  = v;
            out[(long long)rr * HID + col] = di * di * v;   // self-loop init
        }
    }
}

// Final head: out = A[M,64] @ W[64,64] + b  (written straight to d_out)
__global__ void gemm_bias_kernel(const float* __restrict__ A,
                                 const float* __restrict__ W,
                                 const float* __restrict__ bias,
                                 float* __restrict__ out, int M) {
    const int lane = threadIdx.x & 31;
    const int wave = threadIdx.x >> 5;
    const int half = lane >> 4;
    const int m    = lane & 15;
    const int row0 = blockIdx.x * 16;
    const int row  = row0 + m;
    const int rowc = (row < M) ? row : (M - 1);
    const int col  = wave * 16 + m;

    v8f acc = {};
    const float* arow = A + (long long)rowc * HID;
    for (int k = 0; k < HID; k += 4) {
        const int ka = k + half * 2;
        v2f a, b;
        a.x = arow[ka];
        a.y = arow[ka + 1];
        b.x = W[(long long)ka * HID + col];
        b.y = W[(long long)(ka + 1) * HID + col];
        acc = __builtin_amdgcn_wmma_f32_16x16x4_f32(
                  false, a, false, b, (short)0, acc, false, false);
    }

    #pragma unroll
    for (int r = 0; r < 8; ++r) {
        const int rr = row0 + r + 8 * half;
        if (rr < M) out[(long long)rr * HID + col] = acc[r] + bias[col];
    }
}

// -------------------------------------------------------------------
// Edge scatter: out[d] += dinv[s]*dinv[d] * xw[s]
// 16 threads per edge, float4 (4 features) per thread. Both xw and out
// tables (25.6 MB) are L2-resident on the 192 MB L2.
// -------------------------------------------------------------------
__global__ void edge_scatter_kernel(const int* __restrict__ src,
                                    const int* __restrict__ dst,
                                    const float* __restrict__ dinv,
                                    const float* __restrict__ xw,
                                    float* __restrict__ out, int e) {
    const long long idx = (long long)blockIdx.x * blockDim.x + threadIdx.x;
    const long long ed  = idx >> 4;
    if (ed >= e) return;
    const int q = (int)(idx & 15) * 4;
    const int s = src[ed];
    const int d = dst[ed];
    const float coef = dinv[s] * dinv[d];
    const float4 v = *(const float4*)(xw + (long long)s * HID + q);
    float* o = out + (long long)d * HID + q;
    atomicAdd(o + 0, coef * v.x);
    atomicAdd(o + 1, coef * v.y);
    atomicAdd(o + 2, coef * v.z);
    atomicAdd(o + 3, coef * v.w);
}

// h = relu(h + b) in place
__global__ void bias_relu_kernel(float* __restrict__ h, const float* __restrict__ b,
                                 long long total) {
    const long long i = (long long)blockIdx.x * blockDim.x + threadIdx.x;
    if (i < total) {
        const float v = h[i] + b[(int)(i & (HID - 1))];
        h[i] = v > 0.0f ? v : 0.0f;
    }
}

// -------------------------------------------------------------------
// Mean pooling per graph
// -------------------------------------------------------------------
__global__ void zero_kernel(float* __restrict__ p, int n) {
    int i = blockIdx.x * blockDim.x + threadIdx.x;
    if (i < n) p[i] = 0.0f;
}

__global__ void pool_accum_kernel(const float* __restrict__ h,
                                  const int* __restrict__ batch,
                                  float* __restrict__ sums,
                                  float* __restrict__ counts, int n) {
    const long long idx  = (long long)blockIdx.x * blockDim.x + threadIdx.x;
    const long long node = idx >> 6;
    if (node >= n) return;
    const int f = (int)(idx & (HID - 1));
    const int g = batch[node];
    atomicAdd(&sums[(long long)g * HID + f], h[(long long)node * HID + f]);
    if (f == 0) atomicAdd(&counts[g], 1.0f);
}

__global__ void pool_div_kernel(const float* __restrict__ sums,
                                const float* __restrict__ counts,
                                float* __restrict__ pooled) {
    const int i = blockIdx.x * blockDim.x + threadIdx.x;
    if (i < N_GRAPHS * HID) {
        const float c = counts[i >> 6];
        pooled[i] = sums[i] / (c > 1.0f ? c : 1.0f);
    }
}

// ---------------------------------------------------------------------------
extern "C" void kernel_launch(void* const* d_in, const int* in_sizes, int n_in,
                              void* d_out, int out_size, void* d_ws, size_t ws_size,
                              hipStream_t stream) {
    (void)in_sizes; (void)n_in; (void)out_size; (void)ws_size;

    const float* x     = (const float*)d_in[0];
    const int*   eidx  = (const int*)d_in[1];          // [2, E]
    const int*   batch = (const int*)d_in[2];
    const float* W1    = (const float*)d_in[3];
    const float* b1    = (const float*)d_in[4];
    const float* Ws    = (const float*)d_in[5];        // [2, 64, 64]
    const float* bs    = (const float*)d_in[6];        // [2, 64]
    const float* lin_w = (const float*)d_in[7];
    const float* lin_b = (const float*)d_in[8];
    float* out = (float*)d_out;

    const int* src = eidx;
    const int* dst = eidx + N_EDGES;

    // ---- workspace layout (floats) ----
    float* ws     = (float*)d_ws;
    float* dinv   = ws;                                  // N
    float* deg    = dinv + N_NODES;                      // N
    float* xw     = deg + N_NODES;                       // N*64
    float* hA     = xw + (long long)N_NODES * HID;       // N*64
    float* hB     = hA + (long long)N_NODES * HID;       // N*64
    float* sums   = hB + (long long)N_NODES * HID;       // 256*64
    float* counts = sums + N_GRAPHS * HID;               // 256
    float* pooled = counts + N_GRAPHS;                   // 256*64

    const int T = 256;
    const int nodeBlocks  = (N_NODES + T - 1) / T;
    const int edgeBlocks  = (N_EDGES + T - 1) / T;
    const long long featTotal = (long long)N_NODES * HID;
    const int featBlocks  = (int)((featTotal + T - 1) / T);
    const long long escTotal  = (long long)N_EDGES * 16;
    const int escBlocks   = (int)((escTotal + T - 1) / T);
    const int gemmBlocks  = (N_NODES + 15) / 16;

    // 1) degrees + normalization (layer-invariant)
    init_deg_kernel<<<nodeBlocks, T, 0, stream>>>(deg, N_NODES);
    deg_count_kernel<<<edgeBlocks, T, 0, stream>>>(dst, deg, N_EDGES);
    dinv_kernel<<<nodeBlocks, T, 0, stream>>>(deg, dinv, N_NODES);

    // 2) layer 1: x[N,128] @ W1 -> xw; self-loop init into hA; scatter; relu
    gemm_selfloop_kernel<<<gemmBlocks, 128, 0, stream>>>(x, W1, dinv, xw, hA, N_NODES, F_IN);
    edge_scatter_kernel<<<escBlocks, T, 0, stream>>>(src, dst, dinv, xw, hA, N_EDGES);
    bias_relu_kernel<<<featBlocks, T, 0, stream>>>(hA, b1, featTotal);

    // 3) layer 2: hA @ Ws[0] -> hB
    gemm_selfloop_kernel<<<gemmBlocks, 128, 0, stream>>>(hA, Ws, dinv, xw, hB, N_NODES, HID);
    edge_scatter_kernel<<<escBlocks, T, 0, stream>>>(src, dst, dinv, xw, hB, N_EDGES);
    bias_relu_kernel<<<featBlocks, T, 0, stream>>>(hB, bs, featTotal);

    // 4) layer 3: hB @ Ws[1] -> hA
    gemm_selfloop_kernel<<<gemmBlocks, 128, 0, stream>>>(hB, Ws + HID * HID, dinv, xw, hA, N_NODES, HID);
    edge_scatter_kernel<<<escBlocks, T, 0, stream>>>(src, dst, dinv, xw, hA, N_EDGES);
    bias_relu_kernel<<<featBlocks, T, 0, stream>>>(hA, bs + HID, featTotal);

    // 5) mean pool per graph
    zero_kernel<<<(N_GRAPHS * HID + N_GRAPHS + T - 1) / T, T, 0, stream>>>(sums, N_GRAPHS * HID + N_GRAPHS);
    pool_accum_kernel<<<featBlocks, T, 0, stream>>>(hA, batch, sums, counts, N_NODES);
    pool_div_kernel<<<(N_GRAPHS * HID + T - 1) / T, T, 0, stream>>>(sums, counts, pooled);

    // 6) head: pooled[256,64] @ lin_w + lin_b -> d_out
    gemm_bias_kernel<<<N_GRAPHS / 16, 128, 0, stream>>>(pooled, lin_w, lin_b, out, N_GRAPHS);
}